// SpikeFP32Embedding_23407571764103
// MI455X (gfx1250) — compile-verified
//
#include <hip/hip_runtime.h>
#include <hip/hip_bf16.h>

// SpikeFP32Embedding == row gather: out[t, :] = weight_pulse[token_ids[t], :]
//   rows:       16384  (8 batch * 2048 seq)
//   row bytes:  8192   (64 embed * 32 bit-pulses * 4B float)
// Pure bandwidth problem (~256 MB traffic, 0 FLOPs, ~11 us at 23.3 TB/s).
// CDNA5 async data mover: HBM -> LDS -> HBM, 16B per lane per op, ASYNCcnt
// sync, no VGPR data path. Output stores are non-temporal so L2 capacity is
// kept for reusable weight rows instead of write-once output lines.

#define ROW_BYTES 8192u
#define TPB 256

__global__ __launch_bounds__(TPB)
void SpikeFP32Embedding_gather_async(const int* __restrict__ tok,
                                     const float* __restrict__ w,
                                     float* __restrict__ out) {
    __shared__ __align__(16) unsigned char lds_buf[ROW_BYTES];

    const int t = blockIdx.x;              // exactly one block per token

    // Uniform (blockIdx-indexed) -> scalar load + SALU row-base math.
    const unsigned row = (unsigned)tok[t];
    const unsigned tid = threadIdx.x;

    // 32-bit per-lane byte offsets (GVS addressing: SGPR64 base + VGPR i32).
    // Max src offset: 65535*8192 + 8176 < 2^31.  Max dst offset: 2^27.
    unsigned src_off0 = row * ROW_BYTES + tid * 16u;
    unsigned src_off1 = src_off0 + 4096u;
    unsigned dst_off0 = (unsigned)t * ROW_BYTES + tid * 16u;
    unsigned dst_off1 = dst_off0 + 4096u;

    // Low 32 bits of a generic LDS pointer == wave-relative LDS byte offset.
    unsigned lds0 = (unsigned)(unsigned long long)(uintptr_t)&lds_buf[tid * 16u];
    unsigned lds1 = lds0 + 4096u;

    unsigned long long wbase = (unsigned long long)(uintptr_t)w;
    unsigned long long obase = (unsigned long long)(uintptr_t)out;

    // Phase 1: HBM -> LDS (async). 2 x b128 per lane. Default (RT) caching:
    // repeated tokens can re-hit their 8 KB row in L2.
    asm volatile(
        "global_load_async_to_lds_b128 %0, %2, %4\n\t"
        "global_load_async_to_lds_b128 %1, %3, %4\n\t"
        "s_wait_asynccnt 0"
        :
        : "v"(lds0), "v"(lds1), "v"(src_off0), "v"(src_off1), "s"(wbase)
        : "memory");

    // Phase 2: LDS -> HBM (async, non-temporal: write-once output).
    // Each lane stores exactly what it loaded, so the per-wave
    // s_wait_asynccnt above is the only sync required.
    asm volatile(
        "global_store_async_from_lds_b128 %2, %0, %4 th:TH_STORE_NT\n\t"
        "global_store_async_from_lds_b128 %3, %1, %4 th:TH_STORE_NT\n\t"
        "s_wait_asynccnt 0"
        :
        : "v"(lds0), "v"(lds1), "v"(dst_off0), "v"(dst_off1), "s"(obase)
        : "memory");
}

extern "C" void kernel_launch(void* const* d_in, const int* in_sizes, int n_in,
                              void* d_out, int out_size, void* d_ws, size_t ws_size,
                              hipStream_t stream) {
    (void)n_in; (void)d_ws; (void)ws_size; (void)out_size;

    const int*   token_ids = (const int*)d_in[0];      // [8, 2048] int32
    const float* weight    = (const float*)d_in[1];    // [65536, 64, 32] f32
    float*       out       = (float*)d_out;            // [8, 2048, 64, 32] f32

    const int ntok = in_sizes[0];                      // 16384

    dim3 grid((unsigned)ntok);                         // one block per token
    dim3 block(TPB);
    SpikeFP32Embedding_gather_async<<<grid, block, 0, stream>>>(
        token_ids, weight, out);
}